// VMambaBlock_58368605552724
// MI455X (gfx1250) — compile-verified
//
#include <hip/hip_runtime.h>
#include <cmath>

// ---------------------------------------------------------------------------
// VMamba block for MI455X (gfx1250, wave32).
// All GEMMs use V_WMMA_F32_16X16X4_F32 (fp32 reference -> fp32 matrix cores).
// ---------------------------------------------------------------------------

typedef __attribute__((ext_vector_type(2))) float v2f;
typedef __attribute__((ext_vector_type(8))) float v8f;

#define BB     2
#define HH     64
#define WWID   64
#define CIN    96
#define INNER  192
#define LL     4096          // H*W
#define NTOK   (BB*LL)       // 8192 tokens
#define NPAD   160           // 4 * 40 (38 useful cols padded to 40)
#define NST    16
#define RDT    6
#define HID    384

__device__ __forceinline__ float silu_f(float v) { return v / (1.0f + __expf(-v)); }

// ---------------------------------------------------------------------------
// Generic fp32 WMMA GEMM:  C[M x N] = A[M x K] * Wt[N x K]^T  (+ epilogue)
//   mode 0: plain store
//   mode 1: silu(val + bias[col])
//   mode 2: val + resid[row*N+col]
//   mode 3: val + bias[col] + resid[row*N+col]
// 256 threads = 8 waves; wave w computes tile rows [blockIdx.x*128 + w*16, +16),
// cols [blockIdx.y*16, +16). M % 128 == 0, N % 16 == 0, K % 4 == 0 assumed.
// ---------------------------------------------------------------------------
__global__ void gemm_f32_wmma(const float* __restrict__ A,
                              const float* __restrict__ Wt,
                              float* __restrict__ C,
                              const float* __restrict__ bias,
                              const float* __restrict__ resid,
                              int M, int N, int Kd, int mode)
{
    const int lane = threadIdx.x & 31;
    const int wave = threadIdx.x >> 5;
    const int m0 = (blockIdx.x * 8 + wave) * 16;
    const int n0 = blockIdx.y * 16;
    const int r15 = lane & 15;       // row of A-frag / col of B-frag / col of C
    const int hi  = lane >> 4;       // half-wave selects K pair {0,1} vs {2,3}

    v8f acc = {};
    const float* __restrict__ arow = A  + (size_t)(m0 + r15) * Kd;
    const float* __restrict__ brow = Wt + (size_t)(n0 + r15) * Kd;

    for (int k = 0; k < Kd; k += 4) {
        const int kk = k + hi * 2;
        v2f a = *(const v2f*)(arow + kk);   // A[m][kk], A[m][kk+1]
        v2f b = *(const v2f*)(brow + kk);   // B[kk][n] = Wt[n][kk], Wt[n][kk+1]
        acc = __builtin_amdgcn_wmma_f32_16x16x4_f32(
            /*neg_a=*/false, a, /*neg_b=*/false, b,
            /*c_mod=*/(short)0, acc, /*reuse_a=*/false, /*reuse_b=*/false);
    }

    const int col = n0 + r15;
#pragma unroll
    for (int v = 0; v < 8; ++v) {
        const int row = m0 + v + hi * 8;
        float val = acc[v];
        if (mode == 1)      val = silu_f(val + bias[col]);
        else if (mode == 2) val += resid[(size_t)row * N + col];
        else if (mode == 3) val += bias[col] + resid[(size_t)row * N + col];
        C[(size_t)row * N + col] = val;
    }
}

// ---------------------------------------------------------------------------
// Depthwise 3x3 conv (SAME) + bias + SiLU on the xi half of xz.
// xz: (NTOK, 384) tokens row-major; output xc: (NTOK, 192).
// ---------------------------------------------------------------------------
__global__ void dwconv_silu(const float* __restrict__ xz,
                            const float* __restrict__ cw,   // (192,1,3,3)
                            const float* __restrict__ cb,   // (192,)
                            float* __restrict__ xc)
{
    const int idx = blockIdx.x * blockDim.x + threadIdx.x;
    if (idx >= NTOK * INNER) return;
    const int c   = idx % INNER;
    const int tok = idx / INNER;
    const int b = tok >> 12;          // /4096
    const int t = tok & (LL - 1);
    const int h = t >> 6;
    const int w = t & 63;

    float s = cb[c];
#pragma unroll
    for (int dy = 0; dy < 3; ++dy) {
        const int hh = h + dy - 1;
        if (hh < 0 || hh >= HH) continue;
#pragma unroll
        for (int dx = 0; dx < 3; ++dx) {
            const int ww = w + dx - 1;
            if (ww < 0 || ww >= WWID) continue;
            s += xz[((size_t)((b * HH + hh) * WWID + ww)) * 384 + c]
                 * cw[c * 9 + dy * 3 + dx];
        }
    }
    xc[(size_t)tok * INNER + c] = silu_f(s);
}

// ---------------------------------------------------------------------------
// Pack x_proj_w (4, 38, 192) into padded transposed-weight layout Wp (160, 192):
// row (k*40 + c) holds x_proj_w[k][c][:], rows with c in [38,40) are zero.
// ---------------------------------------------------------------------------
__global__ void prep_wp(const float* __restrict__ xpw, float* __restrict__ Wp)
{
    const int idx = blockIdx.x * blockDim.x + threadIdx.x;
    if (idx >= NPAD * INNER) return;
    const int d  = idx % INNER;
    const int nc = idx / INNER;
    const int k  = nc / 40;
    const int c  = nc % 40;
    Wp[idx] = (c < 38) ? xpw[((size_t)k * 38 + c) * INNER + d] : 0.0f;
}

// ---------------------------------------------------------------------------
// Selective scan. One block per (b,k): 8 blocks x 192 threads (one channel d
// per thread). Directions handled by token-index mapping; output stored in
// token order so the merge is a plain sum.
//   P: (NTOK, 160) padded x_dbl;  per (b,k,token) row chunk = P[.. + k*40 ..]
//     [0..5]=dts, [6..21]=B, [22..37]=C
// ---------------------------------------------------------------------------
__global__ void selective_scan(const float* __restrict__ xc,
                               const float* __restrict__ P,
                               const float* __restrict__ dt_w,   // (4,192,6)
                               const float* __restrict__ dt_b,   // (4,192)
                               const float* __restrict__ A_logs, // (4,192,16)
                               const float* __restrict__ Ds,     // (4,192)
                               float* __restrict__ ys)           // (B,4,L,192) token order
{
    __shared__ float sP[40];
    const int b = blockIdx.x >> 2;
    const int k = blockIdx.x & 3;
    const int d = threadIdx.x;

    float dtw[RDT];
#pragma unroll
    for (int r = 0; r < RDT; ++r) dtw[r] = dt_w[((size_t)(k * INNER + d)) * RDT + r];
    const float dbias = dt_b[k * INNER + d];

    float Acoef[NST];
#pragma unroll
    for (int n = 0; n < NST; ++n)
        Acoef[n] = -__expf(A_logs[((size_t)(k * INNER + d)) * NST + n]);
    const float dcoef = Ds[k * INNER + d];

    float hst[NST];
#pragma unroll
    for (int n = 0; n < NST; ++n) hst[n] = 0.0f;

    for (int l = 0; l < LL; ++l) {
        const int lp = (k >= 2) ? (LL - 1 - l) : l;                 // flip dirs
        const int t  = (k & 1) ? ((lp & 63) * 64 + (lp >> 6)) : lp; // col-major dirs

        __syncthreads();                 // protect sP WAR from previous iter
        if (d < 40)
            sP[d] = P[((size_t)(b * LL + t)) * NPAD + k * 40 + d];
        __syncthreads();

        const float xv = xc[((size_t)(b * LL + t)) * INNER + d];

        float raw = dbias;
#pragma unroll
        for (int r = 0; r < RDT; ++r) raw += sP[r] * dtw[r];
        const float delta = (raw > 20.0f) ? raw : log1pf(__expf(raw));  // softplus
        const float dx = delta * xv;

        float y = 0.0f;
#pragma unroll
        for (int n = 0; n < NST; ++n) {
            hst[n] = __expf(delta * Acoef[n]) * hst[n] + dx * sP[6 + n];
            y += hst[n] * sP[22 + n];
        }
        y += dcoef * xv;

        ys[(((size_t)(b * 4 + k)) * LL + t) * INNER + d] = y;
    }
}

// ---------------------------------------------------------------------------
// Cross-merge (sum over 4 directions, already token-ordered) + SiLU(z) gate.
// ---------------------------------------------------------------------------
__global__ void merge_gate(const float* __restrict__ ys,
                           const float* __restrict__ xz,
                           float* __restrict__ ym)
{
    const int idx = blockIdx.x * blockDim.x + threadIdx.x;
    if (idx >= NTOK * INNER) return;
    const int d   = idx % INNER;
    const int tok = idx / INNER;
    const int b = tok >> 12;
    const int t = tok & (LL - 1);

    float s = 0.0f;
#pragma unroll
    for (int k = 0; k < 4; ++k)
        s += ys[(((size_t)(b * 4 + k)) * LL + t) * INNER + d];

    const float z = xz[(size_t)tok * 384 + INNER + d];
    ym[idx] = s * silu_f(z);
}

// ---------------------------------------------------------------------------
extern "C" void kernel_launch(void* const* d_in, const int* in_sizes, int n_in,
                              void* d_out, int out_size, void* d_ws, size_t ws_size,
                              hipStream_t stream)
{
    const float* x          = (const float*)d_in[0];   // (2,64,64,96)
    const float* in_proj_w  = (const float*)d_in[1];   // (384,96)
    const float* conv_w     = (const float*)d_in[2];   // (192,1,3,3)
    const float* conv_b     = (const float*)d_in[3];   // (192,)
    const float* x_proj_w   = (const float*)d_in[4];   // (4,38,192)
    const float* dt_w       = (const float*)d_in[5];   // (4,192,6)
    const float* dt_b       = (const float*)d_in[6];   // (4,192)
    const float* A_logs     = (const float*)d_in[7];   // (4,192,16)
    const float* Ds         = (const float*)d_in[8];   // (4,192)
    const float* out_proj_w = (const float*)d_in[9];   // (96,192)
    const float* mlp_w1     = (const float*)d_in[10];  // (384,96)
    const float* mlp_b1     = (const float*)d_in[11];  // (384,)
    const float* mlp_w2     = (const float*)d_in[12];  // (96,384)
    const float* mlp_b2     = (const float*)d_in[13];  // (96,)
    float* outp = (float*)d_out;                       // (2,64,64,96)

    float* ws = (float*)d_ws;
    float* xz   = ws;                                  // NTOK*384
    float* xc   = xz   + (size_t)NTOK * 384;           // NTOK*192
    float* Wp   = xc   + (size_t)NTOK * INNER;         // 160*192
    float* P    = Wp   + (size_t)NPAD * INNER;         // NTOK*160
    float* ys   = P    + (size_t)NTOK * NPAD;          // NTOK*4*192
    float* ym   = ys   + (size_t)NTOK * 4 * INNER;     // NTOK*192
    float* obuf = ym   + (size_t)NTOK * INNER;         // NTOK*96
    float* hbuf = ys;                                  // alias: ys dead after merge

    // 1) pad/transpose x_proj weights
    prep_wp<<<(NPAD * INNER + 255) / 256, 256, 0, stream>>>(x_proj_w, Wp);

    // 2) in_proj: (8192,96) x (96->384)
    gemm_f32_wmma<<<dim3(NTOK / 128, 384 / 16), 256, 0, stream>>>(
        x, in_proj_w, xz, nullptr, nullptr, NTOK, 384, CIN, 0);

    // 3) depthwise conv + SiLU on xi half
    dwconv_silu<<<(NTOK * INNER + 255) / 256, 256, 0, stream>>>(xz, conv_w, conv_b, xc);

    // 4) x_dbl projection for all 4 directions at once: (8192,192) x (192->160)
    gemm_f32_wmma<<<dim3(NTOK / 128, NPAD / 16), 256, 0, stream>>>(
        xc, Wp, P, nullptr, nullptr, NTOK, NPAD, INNER, 0);

    // 5) selective scan, 4 directions
    selective_scan<<<BB * 4, INNER, 0, stream>>>(xc, P, dt_w, dt_b, A_logs, Ds, ys);

    // 6) cross-merge + gate
    merge_gate<<<(NTOK * INNER + 255) / 256, 256, 0, stream>>>(ys, xz, ym);

    // 7) out_proj + residual x: (8192,192) x (192->96)
    gemm_f32_wmma<<<dim3(NTOK / 128, CIN / 16), 256, 0, stream>>>(
        ym, out_proj_w, obuf, nullptr, x, NTOK, CIN, INNER, 2);

    // 8) mlp1 + bias + SiLU: (8192,96) x (96->384)
    gemm_f32_wmma<<<dim3(NTOK / 128, HID / 16), 256, 0, stream>>>(
        obuf, mlp_w1, hbuf, mlp_b1, nullptr, NTOK, HID, CIN, 1);

    // 9) mlp2 + bias + residual(obuf) -> final output
    gemm_f32_wmma<<<dim3(NTOK / 128, CIN / 16), 256, 0, stream>>>(
        hbuf, mlp_w2, outp, mlp_b2, obuf, NTOK, CIN, HID, 3);
}